// SpatialTransformerNetwork_43319040147995
// MI455X (gfx1250) — compile-verified
//
#include <hip/hip_runtime.h>
#include <math.h>

typedef float v2f __attribute__((ext_vector_type(2)));
typedef float v8f __attribute__((ext_vector_type(8)));

// ---- CDNA5 f32 WMMA: D(16x16) = A(16x4) * B(4x16) + C, full f32 precision ----
__device__ __forceinline__ v8f wmma4(v2f a, v2f b, v8f c) {
  // (neg_a, A, neg_b, B, c_mod, C, reuse_a, reuse_b)
  return __builtin_amdgcn_wmma_f32_16x16x4_f32(false, a, false, b, (short)0, c,
                                               false, false);
}

__device__ __forceinline__ v8f vmax8(v8f a, v8f b) {
  v8f r;
#pragma unroll
  for (int i = 0; i < 8; ++i) r[i] = fmaxf(a[i], b[i]);
  return r;
}

// Implicit-GEMM conv step for one (cy,cx) position over 16 samples:
//   A[m][k] = Ain[m*mStride + base + lut[k]]   (lut zero-padded to nChunks*4)
//   B[k][n] = Bt[k*bStride + n]                (zero-padded rows/cols)
// A-frag per ISA: lane<16 -> M=lane, K = 4c+{0,1}; lane>=16 -> K = 4c+{2,3}.
__device__ __forceinline__ v8f conv_gemm(const float* __restrict__ Ain, int mStride,
                                         int base, const int* __restrict__ lut,
                                         int nChunks, const float* __restrict__ Bt,
                                         int bStride) {
  const int lane = threadIdx.x & 31;
  const float* arow = Ain + (lane & 15) * mStride + base;
  const float* bcol = Bt + (lane & 15);
  const int koff = (lane >> 4) * 2;
  v8f acc = {0.f, 0.f, 0.f, 0.f, 0.f, 0.f, 0.f, 0.f};
  for (int ch = 0; ch < nChunks; ++ch) {
    const int kk = ch * 4 + koff;
    v2f a, b;
    a.x = arow[lut[kk]];
    a.y = arow[lut[kk + 1]];
    b.x = bcol[kk * bStride];
    b.y = bcol[(kk + 1) * bStride];
    acc = wmma4(a, b, acc);
  }
  return acc;
}

// Dense FC GEMM, contiguous K. A rows must be finite out to nChunks*4 (pads zeroed).
__device__ __forceinline__ v8f fc_gemm(const float* __restrict__ Ain, int mStride,
                                       int nChunks, const float* __restrict__ Bt,
                                       int bStride) {
  const int lane = threadIdx.x & 31;
  const float* arow = Ain + (lane & 15) * mStride;
  const float* bcol = Bt + (lane & 15);
  const int koff = (lane >> 4) * 2;
  v8f acc = {0.f, 0.f, 0.f, 0.f, 0.f, 0.f, 0.f, 0.f};
  for (int ch = 0; ch < nChunks; ++ch) {
    const int kk = ch * 4 + koff;
    v2f a, b;
    a.x = arow[kk];
    a.y = arow[kk + 1];
    b.x = bcol[kk * bStride];
    b.y = bcol[(kk + 1) * bStride];
    acc = wmma4(a, b, acc);
  }
  return acc;
}

// FC GEMM with B streamed from global, B[k][col] = gW[col*K + k], masked cols.
__device__ __forceinline__ v8f fc_gemm_gw(const float* __restrict__ Ain, int mStride,
                                          int nChunks, const float* __restrict__ gW,
                                          int K, int colBase, int nCols) {
  const int lane = threadIdx.x & 31;
  const float* arow = Ain + (lane & 15) * mStride;
  const int col = colBase + (lane & 15);
  const float msk = (col < nCols) ? 1.f : 0.f;
  const float* wc = gW + (size_t)(col < nCols ? col : 0) * K;
  const int koff = (lane >> 4) * 2;
  v8f acc = {0.f, 0.f, 0.f, 0.f, 0.f, 0.f, 0.f, 0.f};
  for (int ch = 0; ch < nChunks; ++ch) {
    const int kk = ch * 4 + koff;
    v2f a, b;
    a.x = arow[kk];
    a.y = arow[kk + 1];
    b.x = wc[kk] * msk;
    b.y = wc[kk + 1] * msk;
    acc = wmma4(a, b, acc);
  }
  return acc;
}

// ============================ Kernel 1: localization head ============================
// input tile -> conv1(7x7,1->8)+pool+relu -> conv2(5x5,8->10)+pool+relu
//            -> fc1(90->32,relu) -> fc2(32->6) -> theta
__global__ __launch_bounds__(256) void k_loc(
    const float* __restrict__ inp, const float* __restrict__ w1,
    const float* __restrict__ b1, const float* __restrict__ w2,
    const float* __restrict__ b2, const float* __restrict__ fw1,
    const float* __restrict__ fb1, const float* __restrict__ fw2,
    const float* __restrict__ fb2, float* __restrict__ theta) {
  __shared__ float s_in[16 * 785];       // 28x28 per sample, stride 785
  __shared__ float s_h1[16 * 969];       // (8,11,11) per sample, stride 969
  __shared__ float s_h2[16 * 91 + 8];    // 90 per sample, stride 91, zero-padded
  __shared__ float s_fc1[16 * 33 + 8];   // 32 per sample, stride 33, zero-padded
  __shared__ float s_w1t[52 * 16];       // conv1 Bt, K padded 49->52
  __shared__ float s_w2t[200 * 16];      // conv2 Bt, K=200
  __shared__ float s_f1t[92 * 32];       // fc1 Bt, K padded 90->92, 32 cols
  __shared__ float s_f2t[32 * 16];       // fc2 Bt, K=32
  __shared__ int s_lut1[52];
  __shared__ int s_lut2[200];
  __shared__ float s_b1[8], s_b2[10], s_fb1[32];

  const int tid = threadIdx.x;
  const int n0 = blockIdx.x * 16;

  // ---- stage tables & inputs ----
  {
    const float* g = inp + (size_t)n0 * 784;
    for (int m = 0; m < 16; ++m)
      for (int p = tid; p < 784; p += 256) s_in[m * 785 + p] = g[m * 784 + p];
    for (int i = tid; i < 52 * 16; i += 256) {
      int k = i >> 4, c = i & 15;
      float v = 0.f;
      if (k < 49 && c < 8) v = w1[c * 49 + k];
      s_w1t[i] = v;
    }
    for (int i = tid; i < 200 * 16; i += 256) {
      int k = i >> 4, c = i & 15;
      s_w2t[i] = (c < 10) ? w2[c * 200 + k] : 0.f;
    }
    for (int i = tid; i < 92 * 32; i += 256) {
      int k = i >> 5, c = i & 31;
      s_f1t[i] = (k < 90) ? fw1[c * 90 + k] : 0.f;
    }
    for (int i = tid; i < 32 * 16; i += 256) {
      int k = i >> 4, c = i & 15;
      s_f2t[i] = (c < 6) ? fw2[c * 32 + k] : 0.f;
    }
    for (int i = tid; i < 52; i += 256)
      s_lut1[i] = (i < 49) ? (i / 7) * 28 + (i % 7) : 0;
    for (int i = tid; i < 200; i += 256) {
      int c = i / 25, r = i % 25;
      s_lut2[i] = c * 121 + (r / 5) * 11 + (r % 5);
    }
    for (int i = tid; i < 16 * 91 + 8; i += 256) s_h2[i] = 0.f;
    for (int i = tid; i < 16 * 33 + 8; i += 256) s_fc1[i] = 0.f;
    if (tid < 8) s_b1[tid] = b1[tid];
    if (tid < 10) s_b2[tid] = b2[tid];
    if (tid < 32) s_fb1[tid] = fb1[tid];
  }
  __syncthreads();

  // ---- conv1 + maxpool2 + relu -> s_h1 ----
  {
    const int lane = tid & 31, wv = tid >> 5;
    for (int pos = wv; pos < 121; pos += 8) {
      const int py = pos / 11, px = pos % 11;
      v8f mx = {-1e30f, -1e30f, -1e30f, -1e30f, -1e30f, -1e30f, -1e30f, -1e30f};
#pragma unroll
      for (int dy = 0; dy < 2; ++dy)
#pragma unroll
        for (int dx = 0; dx < 2; ++dx) {
          const int base = (2 * py + dy) * 28 + (2 * px + dx);
          mx = vmax8(mx, conv_gemm(s_in, 785, base, s_lut1, 13, s_w1t, 16));
        }
      const int c = lane & 15, mb = (lane >> 4) * 8;
      if (c < 8) {
        const float bias = s_b1[c];
#pragma unroll
        for (int r = 0; r < 8; ++r)
          s_h1[(mb + r) * 969 + c * 121 + pos] = fmaxf(mx[r] + bias, 0.f);
      }
    }
  }
  __syncthreads();

  // ---- conv2 + maxpool2 + relu -> s_h2 (flattened c*9+py*3+px) ----
  {
    const int lane = tid & 31, wv = tid >> 5;
    for (int pos = wv; pos < 9; pos += 8) {
      const int py = pos / 3, px = pos % 3;
      v8f mx = {-1e30f, -1e30f, -1e30f, -1e30f, -1e30f, -1e30f, -1e30f, -1e30f};
#pragma unroll
      for (int dy = 0; dy < 2; ++dy)
#pragma unroll
        for (int dx = 0; dx < 2; ++dx) {
          const int base = (2 * py + dy) * 11 + (2 * px + dx);
          mx = vmax8(mx, conv_gemm(s_h1, 969, base, s_lut2, 50, s_w2t, 16));
        }
      const int c = lane & 15, mb = (lane >> 4) * 8;
      if (c < 10) {
        const float bias = s_b2[c];
#pragma unroll
        for (int r = 0; r < 8; ++r)
          s_h2[(mb + r) * 91 + c * 9 + pos] = fmaxf(mx[r] + bias, 0.f);
      }
    }
  }
  __syncthreads();

  // ---- fc1: (16x90)x(90x32), relu ----
  {
    const int lane = tid & 31, wv = tid >> 5;
    if (wv < 2) {
      v8f acc = fc_gemm(s_h2, 91, 23, s_f1t + wv * 16, 32);
      const int o = wv * 16 + (lane & 15), mb = (lane >> 4) * 8;
      const float bias = s_fb1[o];
#pragma unroll
      for (int r = 0; r < 8; ++r)
        s_fc1[(mb + r) * 33 + o] = fmaxf(acc[r] + bias, 0.f);
    }
  }
  __syncthreads();

  // ---- fc2: (16x32)x(32x6) -> theta ----
  {
    const int lane = tid & 31, wv = tid >> 5;
    if (wv == 0) {
      v8f acc = fc_gemm(s_fc1, 33, 8, s_f2t, 16);
      const int c = lane & 15, mb = (lane >> 4) * 8;
      if (c < 6) {
        const float bias = fb2[c];
#pragma unroll
        for (int r = 0; r < 8; ++r)
          theta[(size_t)(n0 + mb + r) * 6 + c] = acc[r] + bias;
      }
    }
  }
}

// ============================ Kernel 2: sampler + conv3 ============================
__global__ __launch_bounds__(256) void k_sample_conv3(
    const float* __restrict__ inp, const float* __restrict__ theta,
    const float* __restrict__ w3, const float* __restrict__ b3,
    float* __restrict__ h3) {
  __shared__ float s_in[16 * 785];
  __shared__ float s_samp[16 * 785];
  __shared__ float s_w3t[28 * 16];  // K padded 25->28
  __shared__ int s_lut3[28];
  __shared__ float s_th[16 * 6];
  __shared__ float s_b3[10];

  const int tid = threadIdx.x;
  const int n0 = blockIdx.x * 16;
  {
    const float* g = inp + (size_t)n0 * 784;
    for (int m = 0; m < 16; ++m)
      for (int p = tid; p < 784; p += 256) s_in[m * 785 + p] = g[m * 784 + p];
    for (int i = tid; i < 28 * 16; i += 256) {
      int k = i >> 4, c = i & 15;
      float v = 0.f;
      if (k < 25 && c < 10) v = w3[c * 25 + k];
      s_w3t[i] = v;
    }
    for (int i = tid; i < 28; i += 256)
      s_lut3[i] = (i < 25) ? (i / 5) * 28 + (i % 5) : 0;
    if (tid < 96) s_th[tid] = theta[(size_t)n0 * 6 + tid];
    if (tid < 10) s_b3[tid] = b3[tid];
  }
  __syncthreads();

  // ---- bilinear sampling (identity to reference semantics) ----
  for (int m = 0; m < 16; ++m) {
    const float t0 = s_th[m * 6 + 0], t1 = s_th[m * 6 + 1], t2 = s_th[m * 6 + 2];
    const float t3 = s_th[m * 6 + 3], t4 = s_th[m * 6 + 4], t5 = s_th[m * 6 + 5];
    const float* im = s_in + m * 785;
    for (int p = tid; p < 784; p += 256) {
      const int i = p / 28, j = p % 28;
      const float xl = j * (2.f / 27.f) - 1.f;
      const float yl = i * (2.f / 27.f) - 1.f;
      const float xs = 13.5f * ((t0 * xl + t1 * yl + t2) + 1.f);
      const float ys = 13.5f * ((t3 * xl + t4 * yl + t5) + 1.f);
      const float x0f = floorf(xs), y0f = floorf(ys);
      const float x0d = fminf(fmaxf(x0f, 0.f), 27.f);
      const float x1d = fminf(fmaxf(x0f + 1.f, 0.f), 27.f);
      const float y0d = fminf(fmaxf(y0f, 0.f), 27.f);
      const float y1d = fminf(fmaxf(y0f + 1.f, 0.f), 27.f);
      const int x0 = (int)x0d, x1 = (int)x1d, y0 = (int)y0d, y1 = (int)y1d;
      const float Ia = im[y0 * 28 + x0];
      const float Ib = im[y1 * 28 + x0];
      const float Ic = im[y0 * 28 + x1];
      const float Id = im[y1 * 28 + x1];
      const float wa = (x1d - xs) * (y1d - ys);
      const float wb = (x1d - xs) * (ys - y0d);
      const float wc = (xs - x0d) * (y1d - ys);
      const float wd = (xs - x0d) * (ys - y0d);
      s_samp[m * 785 + p] = wa * Ia + wb * Ib + wc * Ic + wd * Id;
    }
  }
  __syncthreads();

  // ---- conv3 + maxpool2 + relu -> h3 global (N,10,12,12) ----
  {
    const int lane = tid & 31, wv = tid >> 5;
    for (int pos = wv; pos < 144; pos += 8) {
      const int py = pos / 12, px = pos % 12;
      v8f mx = {-1e30f, -1e30f, -1e30f, -1e30f, -1e30f, -1e30f, -1e30f, -1e30f};
#pragma unroll
      for (int dy = 0; dy < 2; ++dy)
#pragma unroll
        for (int dx = 0; dx < 2; ++dx) {
          const int base = (2 * py + dy) * 28 + (2 * px + dx);
          mx = vmax8(mx, conv_gemm(s_samp, 785, base, s_lut3, 7, s_w3t, 16));
        }
      const int c = lane & 15, mb = (lane >> 4) * 8;
      if (c < 10) {
        const float bias = s_b3[c];
#pragma unroll
        for (int r = 0; r < 8; ++r)
          h3[(size_t)(n0 + mb + r) * 1440 + c * 144 + pos] =
              fmaxf(mx[r] + bias, 0.f);
      }
    }
  }
}

// ============================ Kernel 3: classifier head ============================
__global__ __launch_bounds__(256) void k_head(
    const float* __restrict__ h3, const float* __restrict__ w4,
    const float* __restrict__ b4, const float* __restrict__ fw3,
    const float* __restrict__ fb3, const float* __restrict__ fw4,
    const float* __restrict__ fb4, float* __restrict__ out) {
  __shared__ float s_h3[16 * 1441];     // (10,12,12) per sample, stride 1441
  __shared__ float s_h4[16 * 321 + 8];  // 320 per sample, stride 321
  __shared__ float s_h5[16 * 51 + 8];   // 50 per sample, stride 51, zero-padded
  __shared__ float s_o[16 * 12];        // fc4 pre-softmax
  __shared__ float s_w4t[252 * 32];     // conv4 Bt, K padded 250->252, 32 cols
  __shared__ float s_f4t[52 * 16];      // fc4 Bt, K padded 50->52
  __shared__ int s_lut4[252];
  __shared__ float s_b4[20], s_fb3[52];

  const int tid = threadIdx.x;
  const int n0 = blockIdx.x * 16;
  {
    const float* g = h3 + (size_t)n0 * 1440;
    for (int m = 0; m < 16; ++m)
      for (int p = tid; p < 1440; p += 256) s_h3[m * 1441 + p] = g[m * 1440 + p];
    for (int i = tid; i < 252 * 32; i += 256) {
      int k = i >> 5, c = i & 31;
      float v = 0.f;
      if (k < 250 && c < 20) v = w4[c * 250 + k];
      s_w4t[i] = v;
    }
    for (int i = tid; i < 52 * 16; i += 256) {
      int k = i >> 4, c = i & 15;
      float v = 0.f;
      if (k < 50 && c < 10) v = fw4[c * 50 + k];
      s_f4t[i] = v;
    }
    for (int i = tid; i < 252; i += 256) {
      int c = i / 25, r = i % 25;
      s_lut4[i] = (i < 250) ? c * 144 + (r / 5) * 12 + (r % 5) : 0;
    }
    for (int i = tid; i < 16 * 51 + 8; i += 256) s_h5[i] = 0.f;
    if (tid < 20) s_b4[tid] = b4[tid];
    if (tid < 50) s_fb3[tid] = fb3[tid];
  }
  __syncthreads();

  // ---- conv4 + maxpool2 + relu -> s_h4 (flattened c*16+py*4+px) ----
  {
    const int lane = tid & 31, wv = tid >> 5;
    for (int job = wv; job < 32; job += 8) {  // 16 pooled positions x 2 col-tiles
      const int pos = job >> 1, t = job & 1;
      const int py = pos / 4, px = pos % 4;
      v8f mx = {-1e30f, -1e30f, -1e30f, -1e30f, -1e30f, -1e30f, -1e30f, -1e30f};
#pragma unroll
      for (int dy = 0; dy < 2; ++dy)
#pragma unroll
        for (int dx = 0; dx < 2; ++dx) {
          const int base = (2 * py + dy) * 12 + (2 * px + dx);
          mx = vmax8(mx, conv_gemm(s_h3, 1441, base, s_lut4, 63, s_w4t + t * 16, 32));
        }
      const int c = t * 16 + (lane & 15), mb = (lane >> 4) * 8;
      if (c < 20) {
        const float bias = s_b4[c];
#pragma unroll
        for (int r = 0; r < 8; ++r)
          s_h4[(mb + r) * 321 + c * 16 + pos] = fmaxf(mx[r] + bias, 0.f);
      }
    }
  }
  __syncthreads();

  // ---- fc3: (16x320)x(320x50), relu; B streamed from global ----
  {
    const int lane = tid & 31, wv = tid >> 5;
    if (wv < 4) {
      v8f acc = fc_gemm_gw(s_h4, 321, 80, fw3, 320, wv * 16, 50);
      const int o = wv * 16 + (lane & 15), mb = (lane >> 4) * 8;
      if (o < 50) {
        const float bias = s_fb3[o];
#pragma unroll
        for (int r = 0; r < 8; ++r)
          s_h5[(mb + r) * 51 + o] = fmaxf(acc[r] + bias, 0.f);
      }
    }
  }
  __syncthreads();

  // ---- fc4: (16x50)x(50x10) ----
  {
    const int lane = tid & 31, wv = tid >> 5;
    if (wv == 0) {
      v8f acc = fc_gemm(s_h5, 51, 13, s_f4t, 16);
      const int c = lane & 15, mb = (lane >> 4) * 8;
      if (c < 10) {
        const float bias = fb4[c];
#pragma unroll
        for (int r = 0; r < 8; ++r) s_o[(mb + r) * 12 + c] = acc[r] + bias;
      }
    }
  }
  __syncthreads();

  // ---- log_softmax ----
  if (tid < 16) {
    const float* x = s_o + tid * 12;
    float mx = x[0];
#pragma unroll
    for (int j = 1; j < 10; ++j) mx = fmaxf(mx, x[j]);
    float s = 0.f;
#pragma unroll
    for (int j = 0; j < 10; ++j) s += expf(x[j] - mx);
    const float lse = mx + logf(s);
#pragma unroll
    for (int j = 0; j < 10; ++j)
      out[(size_t)(n0 + tid) * 10 + j] = x[j] - lse;
  }
}

extern "C" void kernel_launch(void* const* d_in, const int* in_sizes, int n_in,
                              void* d_out, int out_size, void* d_ws, size_t ws_size,
                              hipStream_t stream) {
  const float* inp = (const float*)d_in[0];
  const float* c1w = (const float*)d_in[1];
  const float* c1b = (const float*)d_in[2];
  const float* c2w = (const float*)d_in[3];
  const float* c2b = (const float*)d_in[4];
  const float* f1w = (const float*)d_in[5];
  const float* f1b = (const float*)d_in[6];
  const float* f2w = (const float*)d_in[7];
  const float* f2b = (const float*)d_in[8];
  const float* c3w = (const float*)d_in[9];
  const float* c3b = (const float*)d_in[10];
  const float* c4w = (const float*)d_in[11];
  const float* c4b = (const float*)d_in[12];
  const float* f3w = (const float*)d_in[13];
  const float* f3b = (const float*)d_in[14];
  const float* f4w = (const float*)d_in[15];
  const float* f4b = (const float*)d_in[16];

  const int N = in_sizes[0] / 784;  // 16384
  const int nb = N / 16;            // 16-sample tiles

  float* theta = (float*)d_ws;
  size_t th_bytes = (((size_t)N * 6 * sizeof(float)) + 255) & ~(size_t)255;
  float* h3 = (float*)((char*)d_ws + th_bytes);  // N*10*12*12 f32

  k_loc<<<nb, 256, 0, stream>>>(inp, c1w, c1b, c2w, c2b, f1w, f1b, f2w, f2b,
                                theta);
  k_sample_conv3<<<nb, 256, 0, stream>>>(inp, theta, c3w, c3b, h3);
  k_head<<<nb, 256, 0, stream>>>(h3, c4w, c4b, f3w, f3b, f4w, f4b,
                                 (float*)d_out);
}